// GraphPoolingLayer_33612414058786
// MI455X (gfx1250) — compile-verified
//
#include <hip/hip_runtime.h>
#include <hip/hip_bf16.h>

typedef float v2f __attribute__((ext_vector_type(2)));
typedef float v8f __attribute__((ext_vector_type(8)));

constexpr int B = 16;
constexpr int N = 20000;
constexpr int D = 256;
constexpr int K = 16000;            // int(N * 0.8)
constexpr int RANK_BLOCK = 1024;
constexpr int RANK_CHUNKS = (N + RANK_BLOCK - 1) / RANK_BLOCK;   // 20

// ---------------------------------------------------------------------------
// Phase 1: scores[b][n] = dot(x[b][n][:], w) + bias  via V_WMMA_F32_16X16X4_F32.
// One wave32 handles a 16-row tile; 64 accumulating WMMAs cover D=256.
// A 16x4 f32 layout (ISA 7.12.2): lanes 0-15 hold row M=lane, K={0,1};
// lanes 16-31 hold row M=lane-16, K={2,3}. B 4x16 layout: VGPR v, lanes 0-15
// -> K=v, lanes 16-31 -> K=v+2; with all 16 columns equal to w, every lane
// loads the same two w values its K-half needs.
// ---------------------------------------------------------------------------
__global__ void score_wmma_kernel(const float* __restrict__ x,
                                  const float* __restrict__ w,
                                  const float* __restrict__ bias,
                                  float* __restrict__ scores) {
    const int wave  = (blockIdx.x * blockDim.x + threadIdx.x) >> 5;
    const int lane  = threadIdx.x & 31;
    const int b     = wave / (N / 16);
    const int n0    = (wave % (N / 16)) * 16;

    const int row   = lane & 15;
    const int khalf = lane >> 4;                 // 0 -> K pair {0,1}, 1 -> {2,3}

    const float* xrow = x + ((size_t)b * N + (n0 + row)) * D;

    v8f acc = {};
#pragma unroll 8
    for (int c = 0; c < D / 4; ++c) {
        const int k0 = c * 4 + 2 * khalf;
        v2f a  = *reinterpret_cast<const v2f*>(xrow + k0);   // 8B aligned
        v2f bw = *reinterpret_cast<const v2f*>(w + k0);
        acc = __builtin_amdgcn_wmma_f32_16x16x4_f32(false, a, false, bw,
                                                    (short)0, acc, false, false);
    }

    // D 16x16 f32 layout: VGPR v, lanes 0-15 hold M=v (N=lane), lanes 16-31
    // hold M=v+8. Column N=0 => lane 0 has rows 0..7, lane 16 has rows 8..15.
    if ((lane & 15) == 0) {
        const float bb = bias[0];
        float* s = scores + (size_t)b * N + n0 + khalf * 8;
#pragma unroll
        for (int v = 0; v < 8; ++v) s[v] = acc[v] + bb;
    }
}

// ---------------------------------------------------------------------------
// Phase 2: exact stable descending rank per element. LDS-staged scores
// (80 KB of CDNA5's 320 KB WGP LDS), inner loop is uniform-address (broadcast)
// float4 LDS reads. rank < K  =>  output row index = rank (ranks are unique).
// ---------------------------------------------------------------------------
__global__ void rank_select_kernel(const float* __restrict__ scores,
                                   int* __restrict__ perm,
                                   float* __restrict__ gate) {
    extern __shared__ float s[];                 // N floats = 80000 bytes
    const int b     = blockIdx.y;
    const int chunk = blockIdx.x;

    const float* sb = scores + (size_t)b * N;
    for (int idx = threadIdx.x; idx < N; idx += RANK_BLOCK) s[idx] = sb[idx];
    __syncthreads();

    const int i = chunk * RANK_BLOCK + threadIdx.x;
    if (i >= N) return;

    const float si = s[i];
    int rank = 0;
    for (int j = 0; j < N; j += 4) {
        const float4 v = *reinterpret_cast<const float4*>(s + j);
        rank += (int)(v.x > si) + (int)((v.x == si) & (j + 0 < i));
        rank += (int)(v.y > si) + (int)((v.y == si) & (j + 1 < i));
        rank += (int)(v.z > si) + (int)((v.z == si) & (j + 2 < i));
        rank += (int)(v.w > si) + (int)((v.w == si) & (j + 3 < i));
    }
    if (rank < K) {
        perm[(size_t)b * K + rank] = i;
        gate[(size_t)b * K + rank] = 1.0f / (1.0f + __expf(-si));
    }
}

// ---------------------------------------------------------------------------
// Phase 3: out[b][r][:] = x[b][perm[b][r]][:] * gate[b][r].
// One wave per row; 2x b128 load/store per lane => 512B per wave per access.
// ---------------------------------------------------------------------------
__global__ void gather_scale_kernel(const float* __restrict__ x,
                                    const int* __restrict__ perm,
                                    const float* __restrict__ gate,
                                    float* __restrict__ out) {
    const int wave = (blockIdx.x * blockDim.x + threadIdx.x) >> 5;
    const int lane = threadIdx.x & 31;
    const int b    = wave / K;
    const int r    = wave - b * K;

    const int   src = perm[(size_t)b * K + r];
    const float g   = gate[(size_t)b * K + r];

    const float4* in4  = reinterpret_cast<const float4*>(x + ((size_t)b * N + src) * D);
    float4*       out4 = reinterpret_cast<float4*>(out + ((size_t)b * K + r) * D);

    float4 p0 = in4[lane];
    float4 p1 = in4[lane + 32];
    p0.x *= g; p0.y *= g; p0.z *= g; p0.w *= g;
    p1.x *= g; p1.y *= g; p1.z *= g; p1.w *= g;
    out4[lane]      = p0;
    out4[lane + 32] = p1;
}

extern "C" void kernel_launch(void* const* d_in, const int* in_sizes, int n_in,
                              void* d_out, int out_size, void* d_ws, size_t ws_size,
                              hipStream_t stream) {
    const float* x    = (const float*)d_in[0];
    const float* w    = (const float*)d_in[1];
    const float* bias = (const float*)d_in[2];
    float* out = (float*)d_out;

    float* scores = (float*)d_ws;                      // B*N   floats
    int*   perm   = (int*)(scores + (size_t)B * N);    // B*K   ints
    float* gate   = (float*)(perm + (size_t)B * K);    // B*K   floats

    // Phase 1: 20000 wave-tiles, 4 waves per 128-thread block.
    {
        const int waves = B * (N / 16);                // 20000
        score_wmma_kernel<<<waves / 4, 128, 0, stream>>>(x, w, bias, scores);
    }
    // Phase 2: 16 batches x 20 chunks, 1024 threads, 80 KB dynamic LDS.
    {
        dim3 grid(RANK_CHUNKS, B);
        rank_select_kernel<<<grid, RANK_BLOCK, N * sizeof(float), stream>>>(scores, perm, gate);
    }
    // Phase 3: 256000 row-waves, 8 waves per 256-thread block.
    {
        const int waves = B * K;                       // 256000
        gather_scale_kernel<<<waves / 8, 256, 0, stream>>>(x, perm, gate, out);
    }
}